// Bandpass_33870112096893
// MI455X (gfx1250) — compile-verified
//
#include <hip/hip_runtime.h>
#include <cstdint>

// ---------------------------------------------------------------------------
// filtfilt of an order-16 (8th-order bandpass) Butterworth IIR, fp64.
//   audio: float32 [128][160000], b,a: float64[17]  ->  out: float64 [128][160000]
//
// MI455X strategy:
//  * fp64 transposed direct-form-II scan. CDNA5 matrix units (WMMA) are
//    fp32-and-below: unusable for a 160k-step order-16 recurrence validated
//    against float64, and the fp32 block-state-space alternative is
//    issue-bound WORSE (~70 instr/sample vs ~36) at this wave count.
//  * Chunked warm-up parallel scan: max pole radius ~0.981 => 1792-sample
//    warm-up decays initial-state error by e^-34 (< fp64 signal precision)
//    => numerically exact. 128 rows x 157 chunks = 628 wave32s (~1/SIMD).
//  * Branch-free segmented inner loops (left-pad / middle / right-pad).
//  * Coefficients staged to LDS via global_load_async_to_lds_b128 +
//    s_wait_asynccnt (CDNA5 ASYNCcnt path); global_prefetch via builtin.
// ---------------------------------------------------------------------------

#define BATCH 128
#define NTAPS 17
#define NST   16            // state length = ntaps-1
#define PADX  (3 * NTAPS)   // 51 (reference's edge-reflection pad)
#define WARM  1792          // warm-up samples (error e^-34, exact to fp64)
#define LCH   1024          // emitted samples per chunk/lane
#define HDR_DOUBLES 64
#define HDR_BYTES   (HDR_DOUBLES * 8)

static __device__ __forceinline__ int imin(int a, int b) { return a < b ? a : b; }
static __device__ __forceinline__ int imax(int a, int b) { return a > b ? a : b; }

// ws header layout (doubles): [0]=b0  [1..16]=b[1..16]  [17..32]=a[1..16]
//                             [33..48]=zi  [49..63]=pad

// --- one-thread setup: normalize coeffs + solve (I - comp^T) zi = B --------
__global__ void bp_setup(const double* __restrict__ b, const double* __restrict__ a,
                         double* __restrict__ hdr)
{
  if (blockIdx.x != 0 || threadIdx.x != 0) return;
  double bn[NTAPS], an[NTAPS];
  double a0 = a[0];
  for (int i = 0; i < NTAPS; ++i) { bn[i] = b[i] / a0; an[i] = a[i] / a0; }

  // M[i][0] = an[i+1] + (i==0);  M[i][j>0] = (i==j) - (i==j-1);  rhs B
  double M[NST][NST + 1];
  for (int i = 0; i < NST; ++i) {
    for (int j = 0; j < NST; ++j) {
      double v = (j == 0) ? an[i + 1] : 0.0;
      if (i == j) v += 1.0;
      if (j > 0 && i == j - 1) v -= 1.0;
      M[i][j] = v;
    }
    M[i][NST] = bn[i + 1] - an[i + 1] * bn[0];
  }
  // Gauss-Jordan with partial pivoting (16x16, fp64)
  for (int c = 0; c < NST; ++c) {
    int p = c; double mx = fabs(M[c][c]);
    for (int r = c + 1; r < NST; ++r) { double v = fabs(M[r][c]); if (v > mx) { mx = v; p = r; } }
    if (p != c) for (int j = c; j <= NST; ++j) { double t = M[c][j]; M[c][j] = M[p][j]; M[p][j] = t; }
    double inv = 1.0 / M[c][c];
    for (int r = 0; r < NST; ++r) {
      if (r == c) continue;
      double f = M[r][c] * inv;
      for (int j = c; j <= NST; ++j) M[r][j] -= f * M[c][j];
    }
  }
  hdr[0] = bn[0];
  for (int i = 0; i < NST; ++i) {
    hdr[1 + i]  = bn[i + 1];
    hdr[17 + i] = an[i + 1];
    hdr[33 + i] = M[i][NST] / M[i][i];   // zi
  }
  for (int i = 49; i < HDR_DOUBLES; ++i) hdr[i] = 0.0;
}

// --- async-stage 512B of coefficients into LDS (CDNA5 ASYNCcnt path) -------
__device__ __forceinline__ void stage_coefs(const double* __restrict__ hdr, double* sc)
{
  if (threadIdx.x < 32) {
    // generic LDS pointer: low 32 bits are the LDS byte offset (aperture form)
    unsigned lds = (unsigned)(uintptr_t)sc + threadIdx.x * 16u;
    unsigned off = threadIdx.x * 16u;
    asm volatile("global_load_async_to_lds_b128 %0, %1, %2\n\t"
                 "s_wait_asynccnt 0x0"
                 :: "v"(lds), "v"(off), "s"(hdr) : "memory");
  }
  __syncthreads();
}

// --- one TDF-II step, fully unrolled, fp64 FMAs -----------------------------
__device__ __forceinline__ double iir_step(double xt, double z[NST], double b0,
                                           const double bt[NST], const double at[NST])
{
  double yt = fma(b0, xt, z[0]);
#pragma unroll
  for (int i = 0; i < NST - 1; ++i)
    z[i] = fma(-at[i], yt, fma(bt[i], xt, z[i + 1]));
  z[NST - 1] = fma(-at[NST - 1], yt, bt[NST - 1] * xt);
  return yt;
}

// --- branch-free segment scans over ext(t) ----------------------------------
// ext(t) = 2*x0 - xr[PADX-t]         , t <  PADX
//        = xr[t-PADX]                , PADX <= t < PADX+T
//        = 2*xl - xr[2T-2+PADX-t]    , t >= PADX+T
template <typename TI, bool STORE>
__device__ __forceinline__ void scan_left(const float* __restrict__ xr, int t0, int t1,
                                          double x02, double z[NST], double b0,
                                          const double bt[NST], const double at[NST],
                                          TI* __restrict__ yw)
{
  for (int t = t0; t < t1; ++t) {
    double yt = iir_step(x02 - (double)xr[PADX - t], z, b0, bt, at);
    if (STORE) yw[t] = (TI)yt;
  }
}
template <typename TI, bool STORE>
__device__ __forceinline__ void scan_mid(const float* __restrict__ xr, int t0, int t1,
                                         double z[NST], double b0,
                                         const double bt[NST], const double at[NST],
                                         TI* __restrict__ yw)
{
  for (int t = t0; t < t1; ++t) {
    double yt = iir_step((double)xr[t - PADX], z, b0, bt, at);
    if (STORE) yw[t] = (TI)yt;
  }
}
template <typename TI, bool STORE>
__device__ __forceinline__ void scan_right(const float* __restrict__ xr, int t0, int t1,
                                           int T, double xl2, double z[NST], double b0,
                                           const double bt[NST], const double at[NST],
                                           TI* __restrict__ yw)
{
  for (int t = t0; t < t1; ++t) {
    double yt = iir_step(xl2 - (double)xr[2 * T - 2 + PADX - t], z, b0, bt, at);
    if (STORE) yw[t] = (TI)yt;
  }
}
template <typename TI, bool STORE>
__device__ __forceinline__ void scan_range(const float* __restrict__ xr, int t0, int t1,
                                           int T, double x02, double xl2, double z[NST],
                                           double b0, const double bt[NST],
                                           const double at[NST], TI* __restrict__ yw)
{
  int aE = imin(t1, PADX);
  if (t0 < aE) scan_left<TI, STORE>(xr, t0, aE, x02, z, b0, bt, at, yw);
  int bS = imax(t0, PADX), bE = imin(t1, PADX + T);
  if (bS < bE) scan_mid<TI, STORE>(xr, bS, bE, z, b0, bt, at, yw);
  int cS = imax(t0, PADX + T);
  if (cS < t1) scan_right<TI, STORE>(xr, cS, t1, T, xl2, z, b0, bt, at, yw);
}

// --- pass 1: forward filter of extended signal -> y1 ------------------------
template <typename TI>
__global__ void bp_pass1(const float* __restrict__ x, const double* __restrict__ hdr,
                         TI* __restrict__ y1, int T, int Text, int chunks)
{
  __shared__ __align__(16) double sc[HDR_DOUBLES];
  stage_coefs(hdr, sc);

  double b0 = sc[0];
  double bt[NST], at[NST];
#pragma unroll
  for (int i = 0; i < NST; ++i) { bt[i] = sc[1 + i]; at[i] = sc[17 + i]; }

  int gid = blockIdx.x * blockDim.x + threadIdx.x;
  if (gid >= BATCH * chunks) return;
  int row = gid / chunks, ch = gid % chunks;

  const float* xr = x + (size_t)row * T;
  double x02 = 2.0 * (double)xr[0];
  double xl2 = 2.0 * (double)xr[T - 1];
  int s  = ch * LCH;
  int e  = imin(s + LCH, Text);
  int p0 = imax(s - WARM, 0);

  __builtin_prefetch(xr + ((p0 >= PADX && p0 < T + PADX) ? (p0 - PADX) : 0), 0, 0);

  double z[NST];
  if (p0 == 0) {
    double e0 = x02 - (double)xr[PADX];      // ext(0): exact init zi * ext[0]
#pragma unroll
    for (int i = 0; i < NST; ++i) z[i] = sc[33 + i] * e0;
  } else {
#pragma unroll
    for (int i = 0; i < NST; ++i) z[i] = 0.0;  // zero-state + warm-up: exact to fp64
  }

  TI* yw = y1 + (size_t)row * Text;
  scan_range<TI, false>(xr, p0, s, T, x02, xl2, z, b0, bt, at, yw);  // warm-up
  scan_range<TI, true >(xr, s,  e, T, x02, xl2, z, b0, bt, at, yw);  // emit
}

// --- pass 2: filter reversed y1, write reversed middle -> out (fp64) --------
template <typename TI>
__global__ void bp_pass2(const TI* __restrict__ y1, const double* __restrict__ hdr,
                         double* __restrict__ out, int T, int Text, int chunks)
{
  __shared__ __align__(16) double sc[HDR_DOUBLES];
  stage_coefs(hdr, sc);

  double b0 = sc[0];
  double bt[NST], at[NST];
#pragma unroll
  for (int i = 0; i < NST; ++i) { bt[i] = sc[1 + i]; at[i] = sc[17 + i]; }

  int gid = blockIdx.x * blockDim.x + threadIdx.x;
  if (gid >= BATCH * chunks) return;
  int row = gid / chunks, ch = gid % chunks;

  const TI* yr  = y1 + (size_t)row * Text;
  double* orow  = out + (size_t)row * T;
  int s  = ch * LCH;
  int e  = imin(s + LCH, Text);
  int p0 = imax(s - WARM, 0);

  __builtin_prefetch(yr + (Text - 1 - p0), 0, 0);

  double z[NST];
  if (p0 == 0) {
    double u0 = (double)yr[Text - 1];        // exact init zi * y[::-1][0]
#pragma unroll
    for (int i = 0; i < NST; ++i) z[i] = sc[33 + i] * u0;
  } else {
#pragma unroll
    for (int i = 0; i < NST; ++i) z[i] = 0.0;
  }

  // warm-up: [p0, s), no stores
  for (int m = p0; m < s; ++m)
    (void)iir_step((double)yr[Text - 1 - m], z, b0, bt, at);

  // emit range split around the store window [PADX, Text-PADX):
  int sS = imax(s, PADX), sE = imin(e, Text - PADX);
  for (int m = s; m < imin(e, PADX); ++m)                 // pre-window (chunk 0 only)
    (void)iir_step((double)yr[Text - 1 - m], z, b0, bt, at);
  for (int m = sS; m < sE; ++m) {                         // step + store, branch-free
    double yt = iir_step((double)yr[Text - 1 - m], z, b0, bt, at);
    orow[Text - 1 - PADX - m] = yt;
  }
  for (int m = imax(s, Text - PADX); m < e; ++m)          // post-window (last chunk)
    (void)iir_step((double)yr[Text - 1 - m], z, b0, bt, at);
}

// ---------------------------------------------------------------------------
extern "C" void kernel_launch(void* const* d_in, const int* in_sizes, int n_in,
                              void* d_out, int out_size, void* d_ws, size_t ws_size,
                              hipStream_t stream)
{
  const float*  audio = (const float*)d_in[0];
  const double* b     = (const double*)d_in[1];
  const double* a     = (const double*)d_in[2];
  double*       out   = (double*)d_out;       // reference output dtype: float64

  int T      = in_sizes[0] / BATCH;           // 160000
  int Text   = T + 2 * PADX;                  // 160102
  int chunks = (Text + LCH - 1) / LCH;        // 157

  double* hdr  = (double*)d_ws;
  char*  y1raw = (char*)d_ws + HDR_BYTES;
  size_t need64 = (size_t)HDR_BYTES + (size_t)BATCH * (size_t)Text * sizeof(double);
  bool   f64ws  = (ws_size >= need64);        // fp64 intermediate if ws allows

  bp_setup<<<1, 1, 0, stream>>>(b, a, hdr);

  int total = BATCH * chunks;                 // 20096 lanes = 628 wave32s
  int blk   = 256;
  int grid  = (total + blk - 1) / blk;

  if (f64ws) {
    double* y1 = (double*)y1raw;
    bp_pass1<double><<<grid, blk, 0, stream>>>(audio, hdr, y1, T, Text, chunks);
    bp_pass2<double><<<grid, blk, 0, stream>>>(y1, hdr, out, T, Text, chunks);
  } else {
    float* y1 = (float*)y1raw;
    bp_pass1<float><<<grid, blk, 0, stream>>>(audio, hdr, y1, T, Text, chunks);
    bp_pass2<float><<<grid, blk, 0, stream>>>(y1, hdr, out, T, Text, chunks);
  }
}